// MoebiusTransformer_73684458930841
// MI455X (gfx1250) — compile-verified
//
#include <hip/hip_runtime.h>

#ifndef __has_builtin
#define __has_builtin(x) 0
#endif

#if __has_builtin(__builtin_amdgcn_global_load_async_to_lds_b128)
#define MOEB_ASYNC 1
#else
#define MOEB_ASYNC 0
#endif

namespace {
constexpr int kBatch = 4096;
constexpr int kVec = 2048;            // vectors per batch row
constexpr int kNF = 6144;             // kVec * 3
constexpr int kTPB = 256;             // threads per block (8 wave32s)
constexpr int kVPT = kVec / kTPB;     // 8 vectors per thread
constexpr int kF4PT = kNF / 4 / kTPB; // 6 float4 loads per thread per array
constexpr float kMaxRadius = 0.99f;
}

#if MOEB_ASYNC
typedef int v4i __attribute__((__vector_size__(16)));
typedef __attribute__((address_space(1))) v4i gv4i;   // global int4
typedef __attribute__((address_space(3))) v4i lv4i;   // LDS int4
#endif

// raw hardware transcendentals (~1 ulp, no refinement sequences)
__device__ __forceinline__ float hw_rcp(float v)  { return __builtin_amdgcn_rcpf(v); }
__device__ __forceinline__ float hw_sqrt(float v) { return __builtin_amdgcn_sqrtf(v); }

__global__ __launch_bounds__(kTPB) void moebius_fused(
    const float* __restrict__ x, const float* __restrict__ w,
    float* __restrict__ out)
{
    __shared__ __align__(16) float sx[kNF];   // x row; reused as y staging
    __shared__ __align__(16) float sw[kNF];   // w row
    __shared__ float sred[kTPB / 32];

    const int b = blockIdx.x;
    const int tid = threadIdx.x;

    const float4* __restrict__ x4 = (const float4*)(x + (size_t)b * kNF);
    const float4* __restrict__ w4 = (const float4*)(w + (size_t)b * kNF);
    float4* sx4 = (float4*)sx;
    float4* sw4 = (float4*)sw;

    // ---- Stage both rows into LDS, 128-bit coalesced async copies (CDNA5) ----
#if MOEB_ASYNC
    #pragma unroll
    for (int i = 0; i < kF4PT; ++i) {
        int idx = i * kTPB + tid;
        __builtin_amdgcn_global_load_async_to_lds_b128(
            (gv4i*)(x4 + idx), (lv4i*)(sx4 + idx), 0, 0);
        __builtin_amdgcn_global_load_async_to_lds_b128(
            (gv4i*)(w4 + idx), (lv4i*)(sw4 + idx), 0, 0);
    }
#if __has_builtin(__builtin_amdgcn_s_wait_asynccnt)
    __builtin_amdgcn_s_wait_asynccnt(0);
#else
    asm volatile("s_wait_asynccnt 0" ::: "memory");
#endif
#else
    #pragma unroll
    for (int i = 0; i < kF4PT; ++i) {
        int idx = i * kTPB + tid;
        sx4[idx] = x4[idx];
        sw4[idx] = w4[idx];
    }
#endif
    __syncthreads();

    // ---- Per-vector Moebius map + closed-form log|det J| ----
    // J collapses to a*I + d*p^T + x*q^T (rank-2 update of scaled identity):
    //   det(J) = a^2 * (1 + (t/|x|^2) * (a + 1 - 2t/|d|^2)),  t = d.x, a = num/|d|^2
    float logdet_acc = 0.0f;
    #pragma unroll
    for (int k = 0; k < kVPT; ++k) {
        const int v = k * kTPB + tid;   // interleaved: stride-3 LDS words, conflict-free
        const int o = 3 * v;
        float x0 = sx[o + 0], x1 = sx[o + 1], x2 = sx[o + 2];
        float w0 = sw[o + 0], w1 = sw[o + 1], w2 = sw[o + 2];

        float xn2 = fmaf(x0, x0, fmaf(x1, x1, x2 * x2));
        float wn0 = hw_sqrt(fmaf(w0, w0, fmaf(w1, w1, w2 * w2)));
        float xn  = hw_sqrt(xn2);
        float inv_xn2 = hw_rcp(xn2);

        float r = kMaxRadius * xn * hw_rcp(1.0f + wn0);   // rescaling
        float ws0 = r * w0, ws1 = r * w1, ws2 = r * w2;
        float wn  = r * wn0;
        float num = fmaf(-wn, wn, xn2);                   // x_norm^2 - w_norm^2 (> 0)

        float d0 = x0 - ws0, d1 = x1 - ws1, d2 = x2 - ws2;
        float dn2 = fmaf(d0, d0, fmaf(d1, d1, d2 * d2));
        float inv_dn2 = hw_rcp(dn2);
        float a = num * inv_dn2;

        // y = a*diff - w_scaled
        float y0 = fmaf(a, d0, -ws0);
        float y1 = fmaf(a, d1, -ws1);
        float y2 = fmaf(a, d2, -ws2);

        float t = fmaf(d0, x0, fmaf(d1, x1, d2 * x2));
        float g = fmaf(t * inv_xn2, fmaf(-2.0f * t, inv_dn2, a + 1.0f), 1.0f);
        logdet_acc += __logf(fabsf(a * a * g));

        // stage y in place of x (this slot is owned exclusively by this thread)
        sx[o + 0] = y0; sx[o + 1] = y1; sx[o + 2] = y2;
    }

    // ---- wave32 reduction of log-det, then cross-wave via LDS ----
    float s = logdet_acc;
    #pragma unroll
    for (int off = 16; off > 0; off >>= 1)
        s += __shfl_down(s, off, 32);
    if ((tid & 31) == 0) sred[tid >> 5] = s;
    __syncthreads();   // also fences the y staging writes in sx

    // ---- write y back, 128-bit coalesced ----
    float4* __restrict__ y4 = (float4*)(out + (size_t)b * kNF);
    #pragma unroll
    for (int i = 0; i < kF4PT; ++i) {
        int idx = i * kTPB + tid;
        y4[idx] = sx4[idx];
    }

    if (tid == 0) {
        float tot = 0.0f;
        #pragma unroll
        for (int i = 0; i < kTPB / 32; ++i) tot += sred[i];
        out[(size_t)kBatch * kNF + b] = tot;   // log_det_J[b] after the y block
    }
}

extern "C" void kernel_launch(void* const* d_in, const int* in_sizes, int n_in,
                              void* d_out, int out_size, void* d_ws, size_t ws_size,
                              hipStream_t stream) {
    const float* x = (const float*)d_in[0];   // (4096, 6144) f32
    const float* w = (const float*)d_in[1];   // (4096, 1, 6144) f32, same flat layout
    float* out = (float*)d_out;               // y (4096*6144) then log_det_J (4096)
    moebius_fused<<<dim3(kBatch), dim3(kTPB), 0, stream>>>(x, w, out);
}